// BipartPool_74560632258836
// MI455X (gfx1250) — compile-verified
//
#include <hip/hip_runtime.h>
#include <hip/hip_bf16.h>

typedef __attribute__((ext_vector_type(2))) float v2f;
typedef __attribute__((ext_vector_type(8))) float v8f;

#define N_NODES      32768
#define B_GRAPHS     32
#define NODES_PER_G  1024
#define RATIO        16
#define C_IN         64
#define HEADS        4
#define HC           256           // HEADS * C_IN
#define T_CENT       512           // B_GRAPHS * RATIO
#define NEG_SLOPE    0.2f
#define KSPLIT       8             // K-chunks for the pooling reduction
#define KCHUNK       (NODES_PER_G / KSPLIT)   // 128

// ---------------------------------------------------------------------------
// CDNA5 async global->LDS copy helpers (ASYNCcnt-tracked, bypasses VGPRs).
// ---------------------------------------------------------------------------
__device__ __forceinline__ unsigned lds_addr_of(const void* p) {
    // Flat address of a __shared__ object: high 32 bits are the LDS aperture,
    // low 32 bits are the wave-relative LDS byte address (ISA 10.2).
    return (unsigned)(unsigned long long)p;
}
__device__ __forceinline__ void async_copy_b128(unsigned lds, const float* g) {
    asm volatile("global_load_async_to_lds_b128 %0, %1, off"
                 :: "v"(lds), "v"(g) : "memory");
}
__device__ __forceinline__ void wait_async0() {
    asm volatile("s_wait_asynccnt 0x0" ::: "memory");
}

// ---------------------------------------------------------------------------
// Kernel 1: x_l = x @ W_l + b_l   ([32768 x 64] @ [64 x 256])
// One wave computes a 16-row band across all 16 N-tiles. WMMA F32 16x16x4.
// W_l (64 KB) staged into LDS with async copies overlapped with A-frag loads.
// ---------------------------------------------------------------------------
__global__ __launch_bounds__(256)
void xl_gemm_kernel(const float* __restrict__ x, const float* __restrict__ W,
                    const float* __restrict__ b, float* __restrict__ xl)
{
    __shared__ float lw[C_IN * HC];           // 64 KB
    const int tid = threadIdx.x;

    // 64 KB / (256 threads * 16 B) = 16 async b128 chunks per thread
    #pragma unroll
    for (int j = 0; j < 16; ++j) {
        const int idx = (j * 256 + tid) * 4;
        async_copy_b128(lds_addr_of(&lw[idx]), W + idx);
    }

    const int wave = tid >> 5;
    const int lane = tid & 31;
    const int lo   = lane & 15;
    const int hi   = lane >> 4;
    const int m0   = (blockIdx.x * 8 + wave) * 16;

    // Load A fragments for the full K=64 strip while the async copy runs.
    v2f a[16];
    const float* xrow = x + (size_t)(m0 + lo) * C_IN;
    #pragma unroll
    for (int k = 0; k < 16; ++k) {
        const int col = 4 * k + 2 * hi;
        a[k].x = xrow[col];
        a[k].y = xrow[col + 1];
    }

    wait_async0();
    __syncthreads();

    for (int n = 0; n < 16; ++n) {
        const int nc = n * 16 + lo;
        const float bv = b[nc];
        v8f c = { bv, bv, bv, bv, bv, bv, bv, bv };
        #pragma unroll
        for (int k = 0; k < 16; ++k) {
            const int r0 = 4 * k + 2 * hi;
            v2f bf;
            bf.x = lw[r0 * HC + nc];
            bf.y = lw[(r0 + 1) * HC + nc];
            c = __builtin_amdgcn_wmma_f32_16x16x4_f32(false, a[k], false, bf,
                                                      (short)0, c, false, false);
        }
        float* orow = xl + nc;
        #pragma unroll
        for (int v = 0; v < 8; ++v)
            orow[(size_t)(m0 + v + 8 * hi) * HC] = c[v];
    }
}

// ---------------------------------------------------------------------------
// Kernel 2: x_r_base = xcent_base @ W_r + b_r   ([16 x 64] @ [64 x 256], tiny)
// ---------------------------------------------------------------------------
__global__ __launch_bounds__(256)
void xr_kernel(const float* __restrict__ xc, const float* __restrict__ W,
               const float* __restrict__ b, float* __restrict__ xr)
{
    const int idx = blockIdx.x * 256 + threadIdx.x;   // 16*256 = 4096 total
    const int r = idx >> 8;
    const int col = idx & 255;
    float acc = b[col];
    #pragma unroll 8
    for (int k = 0; k < C_IN; ++k)
        acc += xc[r * C_IN + k] * W[k * HC + col];
    xr[idx] = acc;
}

// ---------------------------------------------------------------------------
// Kernel 3: logit[i, r, h] = sum_c lrelu(xl[i,h,c] + xr[r,h,c]) * att[h,c]
// One thread per (node, r, h). xr (16 KB) + att (1 KB) async-staged in LDS.
// ---------------------------------------------------------------------------
__global__ __launch_bounds__(256)
void logit_kernel(const float* __restrict__ xl, const float* __restrict__ xr,
                  const float* __restrict__ att, float* __restrict__ logit)
{
    __shared__ float lxr[RATIO * HC];   // 16 KB
    __shared__ float latt[HC];          // 1 KB
    const int tid = threadIdx.x;
    #pragma unroll
    for (int j = 0; j < 4; ++j) {
        const int idx = (j * 256 + tid) * 4;
        async_copy_b128(lds_addr_of(&lxr[idx]), xr + idx);
    }
    if (tid < 64)
        async_copy_b128(lds_addr_of(&latt[tid * 4]), att + tid * 4);
    wait_async0();
    __syncthreads();

    const int gidx = blockIdx.x * 256 + tid;  // N_NODES*64 elements
    const int i  = gidx >> 6;
    const int rh = gidx & 63;
    const int r = rh >> 2, h = rh & 3;

    const float* xli = xl + (size_t)i * HC + h * C_IN;
    const float* xrr = lxr + r * HC + h * C_IN;
    const float* ah  = latt + h * C_IN;
    float acc = 0.f;
    #pragma unroll 8
    for (int c = 0; c < C_IN; ++c) {
        float z = xli[c] + xrr[c];
        z = (z >= 0.f) ? z : NEG_SLOPE * z;
        acc += z * ah[c];
    }
    logit[gidx] = acc;
}

// ---------------------------------------------------------------------------
// Kernel 4: segment softmax stats per (g, r, h).
// 2048 blocks, 256 threads reduce 1024 logits. rden = 0.25/den folds the
// mean-over-heads into the reciprocal.
// ---------------------------------------------------------------------------
__global__ __launch_bounds__(256)
void softmax_stats_kernel(const float* __restrict__ logit,
                          float* __restrict__ mout, float* __restrict__ rden)
{
    __shared__ float red[256];
    const int seg = blockIdx.x;           // g*64 + rh
    const int g = seg >> 6;
    const int rh = seg & 63;
    const int tid = threadIdx.x;
    const float* base = logit + (size_t)g * NODES_PER_G * 64 + rh;

    float v[4];
    float mx = -1e30f;
    #pragma unroll
    for (int j = 0; j < 4; ++j) {
        v[j] = base[(size_t)(tid + 256 * j) * 64];
        mx = fmaxf(mx, v[j]);
    }
    red[tid] = mx;
    __syncthreads();
    for (int s = 128; s > 0; s >>= 1) {
        if (tid < s) red[tid] = fmaxf(red[tid], red[tid + s]);
        __syncthreads();
    }
    const float m = red[0];
    __syncthreads();

    float sum = 0.f;
    #pragma unroll
    for (int j = 0; j < 4; ++j) sum += __expf(v[j] - m);
    red[tid] = sum;
    __syncthreads();
    for (int s = 128; s > 0; s >>= 1) {
        if (tid < s) red[tid] += red[tid + s];
        __syncthreads();
    }
    if (tid == 0) {
        mout[seg] = m;
        rden[seg] = 0.25f / red[0];
    }
}

// ---------------------------------------------------------------------------
// Kernel 5: partial pool. Block = (g, 16-col tile, K-chunk); 4 waves = 4 heads.
// Each wave does a KCHUNK WMMA-f32 reduction with A = exp(logit-m)*rden
// generated on the fly; heads combined through LDS; deterministic per-chunk
// partials written to ws (summed by the finalize kernel — no float atomics).
// ---------------------------------------------------------------------------
__global__ __launch_bounds__(128)
void pool_partial_kernel(const float* __restrict__ xl, const float* __restrict__ logit,
                         const float* __restrict__ m, const float* __restrict__ rden,
                         float* __restrict__ xpart)
{
    __shared__ float red[HEADS][256];
    const int kc    = blockIdx.x & (KSPLIT - 1);
    const int ntile = (blockIdx.x >> 3) & 3;
    const int g     = blockIdx.x >> 5;
    const int n0    = ntile * 16;
    const int wave  = threadIdx.x >> 5;   // = head
    const int lane  = threadIdx.x & 31;
    const int lo = lane & 15;
    const int hi = lane >> 4;
    const int h  = wave;

    const float mm = m[g * 64 + lo * 4 + h];
    const float rd = rden[g * 64 + lo * 4 + h];
    const size_t i0 = (size_t)g * NODES_PER_G + kc * KCHUNK;
    const float* lg = logit + i0 * 64 + lo * 4 + h;
    const float* xb = xl + i0 * HC + h * C_IN + n0 + lo;

    v8f acc = {0.f, 0.f, 0.f, 0.f, 0.f, 0.f, 0.f, 0.f};
    for (int k0 = 0; k0 < KCHUNK; k0 += 4) {
        const int j0 = k0 + 2 * hi;
        v2f a, bf;
        a.x = __expf(lg[(size_t)j0 * 64] - mm) * rd;
        a.y = __expf(lg[(size_t)(j0 + 1) * 64] - mm) * rd;
        bf.x = xb[(size_t)j0 * HC];
        bf.y = xb[(size_t)(j0 + 1) * HC];
        acc = __builtin_amdgcn_wmma_f32_16x16x4_f32(false, a, false, bf,
                                                    (short)0, acc, false, false);
    }

    #pragma unroll
    for (int v = 0; v < 8; ++v) red[wave][v * 32 + lane] = acc[v];
    __syncthreads();

    if (wave == 0) {
        #pragma unroll
        for (int v = 0; v < 8; ++v) {
            const int idx = v * 32 + lane;
            const float s = red[0][idx] + red[1][idx] + red[2][idx] + red[3][idx];
            const int r = v + 8 * hi;
            const int t = g * RATIO + r;
            xpart[(size_t)kc * (T_CENT * C_IN) + (size_t)t * C_IN + n0 + lo] = s;
        }
    }
}

// ---------------------------------------------------------------------------
// Kernel 6: finalize. Sum the KSPLIT partials + bias -> xcent; emit batchcent.
// d_out layout: [512*64] xcent floats, then [512] batchcent values.
// ---------------------------------------------------------------------------
__global__ __launch_bounds__(256)
void finalize_kernel(const float* __restrict__ xpart, const float* __restrict__ bias,
                     float* __restrict__ out)
{
    const int idx = blockIdx.x * 256 + threadIdx.x;
    if (idx < T_CENT * C_IN) {
        float s = bias[idx & 63];
        #pragma unroll
        for (int kc = 0; kc < KSPLIT; ++kc)
            s += xpart[(size_t)kc * (T_CENT * C_IN) + idx];
        out[idx] = s;
    } else if (idx < T_CENT * C_IN + T_CENT) {
        const int t = idx - T_CENT * C_IN;
        out[idx] = (float)(t >> 4);   // repeat(arange(32), 16)
    }
}

// ---------------------------------------------------------------------------
extern "C" void kernel_launch(void* const* d_in, const int* in_sizes, int n_in,
                              void* d_out, int out_size, void* d_ws, size_t ws_size,
                              hipStream_t stream) {
    (void)in_sizes; (void)n_in; (void)out_size; (void)ws_size;

    const float* x          = (const float*)d_in[0];
    // d_in[1] edge_index (ignored by reference forward), d_in[2] batch (implicit: sorted, 1024/graph)
    const float* xcent_base = (const float*)d_in[3];
    const float* W_l        = (const float*)d_in[4];
    const float* b_l        = (const float*)d_in[5];
    const float* W_r        = (const float*)d_in[6];
    const float* b_r        = (const float*)d_in[7];
    const float* att        = (const float*)d_in[8];
    const float* bias       = (const float*)d_in[9];
    float* out = (float*)d_out;

    float* xl    = (float*)d_ws;                         // 32768*256 = 32 MB
    float* xr    = xl + (size_t)N_NODES * HC;            // 16*256
    float* logit = xr + RATIO * HC;                      // 32768*64 = 8 MB
    float* mbuf  = logit + (size_t)N_NODES * 64;         // 2048
    float* rden  = mbuf + B_GRAPHS * 64;                 // 2048
    float* xpart = rden + B_GRAPHS * 64;                 // 8*512*64 = 1 MB

    xl_gemm_kernel<<<N_NODES / (16 * 8), 256, 0, stream>>>(x, W_l, b_l, xl);
    xr_kernel<<<(RATIO * HC) / 256, 256, 0, stream>>>(xcent_base, W_r, b_r, xr);
    logit_kernel<<<(N_NODES * 64) / 256, 256, 0, stream>>>(xl, xr, att, logit);
    softmax_stats_kernel<<<B_GRAPHS * 64, 256, 0, stream>>>(logit, mbuf, rden);
    pool_partial_kernel<<<B_GRAPHS * 4 * KSPLIT, 128, 0, stream>>>(xl, logit, mbuf, rden, xpart);
    finalize_kernel<<<(T_CENT * C_IN + T_CENT + 255) / 256, 256, 0, stream>>>(xpart, bias, out);
}